// QKernel_68839735820835
// MI455X (gfx1250) — compile-verified
//
#include <hip/hip_runtime.h>
#include <math.h>

// ---------------------------------------------------------------------------
// Quantum kernel ridge evaluation, simplified:
//   out[n] = sum_m alpha[m] * ( dotA(n,m) * dotB(n,m) )^2
// dotA/dotB are 64-dim dot products of half product-states (the CNOT ring is
// the same permutation on both sides of the inner product -> cancels).
// Dot products via V_WMMA_F32_16X16X4_F32 (full f32 precision, f32 accum).
//
// K-layout trick: the reduction over K=64 is order-invariant, so fragment
// slot (k-step k, vgpr v, lane-half kh) is mapped to column kh*32 + 2k + v.
// Each lane's 16 fragments are then one contiguous 128-byte run of its row:
// 8 global_load_b128 per factor instead of 16 dependent b64 loads.
// ---------------------------------------------------------------------------

typedef __attribute__((ext_vector_type(2))) float v2f;
typedef __attribute__((ext_vector_type(4))) float v4f;
typedef __attribute__((ext_vector_type(8))) float v8f;

#define NQ 12
#define HD 64   // 2^6, half-state dimension

// Build half product-state tables.
// A[row][i] = prod_{q=0..5} (bit_{5-q}(i) ? sin(ang[row][q]/2)   : cos(ang[row][q]/2))
// B[row][i] = prod_{q=0..5} (bit_{5-q}(i) ? sin(ang[row][6+q]/2) : cos(ang[row][6+q]/2))
__global__ void build_halves_kernel(const float* __restrict__ ang, int rows,
                                    float* __restrict__ A, float* __restrict__ B) {
  int t = blockIdx.x * blockDim.x + threadIdx.x;
  if (t >= rows * HD) return;
  int row = t >> 6;
  int i   = t & 63;
  float pa = 1.0f, pb = 1.0f;
#pragma unroll
  for (int q = 0; q < 6; ++q) {
    int bit = (i >> (5 - q)) & 1;
    float ha = 0.5f * ang[row * NQ + q];
    float hb = 0.5f * ang[row * NQ + 6 + q];
    float sa, ca, sb, cb;
    sincosf(ha, &sa, &ca);
    sincosf(hb, &sb, &cb);
    pa *= bit ? sa : ca;
    pb *= bit ? sb : cb;
  }
  A[row * HD + i] = pa;
  B[row * HD + i] = pb;
}

__global__ void zero_kernel(float* __restrict__ p, int n) {
  int t = blockIdx.x * blockDim.x + threadIdx.x;
  if (t < n) p[t] = 0.0f;
}

__device__ __forceinline__ v2f lo2(v4f x) { v2f r = {x.x, x.y}; return r; }
__device__ __forceinline__ v2f hi2(v4f x) { v2f r = {x.z, x.w}; return r; }

// One wave (32 threads) per block.
// blockIdx.x : n-tile (16 rows of x)
// blockIdx.y : m-chunk (64 training rows = 4 WMMA tiles)
__global__ void __launch_bounds__(32)
qkernel_wmma(const float* __restrict__ Ax, const float* __restrict__ Bx,
             const float* __restrict__ At, const float* __restrict__ Bt,
             const float* __restrict__ alpha, float* __restrict__ out) {
  const int lane = threadIdx.x;   // 0..31
  const int r    = lane & 15;     // row (A) / col (B) within 16x16 tile
  const int kh   = lane >> 4;     // lane-half: columns kh*32 .. kh*32+31
  const int n0     = blockIdx.x * 16;
  const int m0base = blockIdx.y * 64;

  // Hoist x-side fragments (reused across all 4 m-tiles): 2 x 8 x b128.
  v4f xa[8], xb[8];
  {
    const v4f* axp = (const v4f*)(Ax + (size_t)(n0 + r) * HD + kh * 32);
    const v4f* bxp = (const v4f*)(Bx + (size_t)(n0 + r) * HD + kh * 32);
#pragma unroll
    for (int j = 0; j < 8; ++j) { xa[j] = axp[j]; xb[j] = bxp[j]; }
  }

  // Per-lane row accumulators: C VGPR v holds row (v + 8*kh), col r.
  float total[8];
#pragma unroll
  for (int v = 0; v < 8; ++v) total[v] = 0.0f;

#pragma unroll
  for (int mt = 0; mt < 4; ++mt) {
    const int m0 = m0base + mt * 16;

    // Preload all t-side fragments for this tile (16 x b128, clause-able).
    v4f ta[8], tb[8];
    {
      const v4f* atp = (const v4f*)(At + (size_t)(m0 + r) * HD + kh * 32);
      const v4f* btp = (const v4f*)(Bt + (size_t)(m0 + r) * HD + kh * 32);
#pragma unroll
      for (int j = 0; j < 8; ++j) { ta[j] = atp[j]; tb[j] = btp[j]; }
    }

    v8f accA = {};
    v8f accB = {};
#pragma unroll
    for (int j = 0; j < 8; ++j) {
      accA = __builtin_amdgcn_wmma_f32_16x16x4_f32(
          false, lo2(xa[j]), false, lo2(ta[j]), (short)0, accA, false, false);
      accA = __builtin_amdgcn_wmma_f32_16x16x4_f32(
          false, hi2(xa[j]), false, hi2(ta[j]), (short)0, accA, false, false);
      accB = __builtin_amdgcn_wmma_f32_16x16x4_f32(
          false, lo2(xb[j]), false, lo2(tb[j]), (short)0, accB, false, false);
      accB = __builtin_amdgcn_wmma_f32_16x16x4_f32(
          false, hi2(xb[j]), false, hi2(tb[j]), (short)0, accB, false, false);
    }

    const float al = alpha[m0 + r];
#pragma unroll
    for (int v = 0; v < 8; ++v) {
      float o = accA[v] * accB[v];   // overlap = dotA * dotB
      total[v] += o * o * al;        // K = overlap^2, alpha-weighted
    }
  }

  // Reduce over the 16 columns (lanes within each 16-lane half), then one
  // atomicAdd per output row from lane 0 of each half.
#pragma unroll
  for (int v = 0; v < 8; ++v) {
    float s = total[v];
    s += __shfl_xor(s, 1, 32);
    s += __shfl_xor(s, 2, 32);
    s += __shfl_xor(s, 4, 32);
    s += __shfl_xor(s, 8, 32);
    if (r == 0) atomicAdd(out + n0 + kh * 8 + v, s);
  }
}

extern "C" void kernel_launch(void* const* d_in, const int* in_sizes, int n_in,
                              void* d_out, int out_size, void* d_ws, size_t ws_size,
                              hipStream_t stream) {
  const float* x     = (const float*)d_in[0];   // (512, 12)
  const float* Xtr   = (const float*)d_in[1];   // (4096, 12)
  const float* alpha = (const float*)d_in[2];   // (4096,)
  float* out = (float*)d_out;                   // (512,)

  const int nx = in_sizes[0] / NQ;  // 512
  const int nt = in_sizes[1] / NQ;  // 4096

  // Workspace layout: half-state tables, ~2.36 MB total (L2 resident).
  float* Ax = (float*)d_ws;
  float* Bx = Ax + (size_t)nx * HD;
  float* At = Bx + (size_t)nx * HD;
  float* Bt = At + (size_t)nt * HD;

  const int T = 256;
  build_halves_kernel<<<(nx * HD + T - 1) / T, T, 0, stream>>>(x,   nx, Ax, Bx);
  build_halves_kernel<<<(nt * HD + T - 1) / T, T, 0, stream>>>(Xtr, nt, At, Bt);
  zero_kernel<<<(out_size + T - 1) / T, T, 0, stream>>>(out, out_size);

  dim3 grid(nx / 16, nt / 64);   // 32 x 64 = 2048 single-wave blocks
  qkernel_wmma<<<grid, 32, 0, stream>>>(Ax, Bx, At, Bt, alpha, out);
}